// GCNlayer_51084341019209
// MI455X (gfx1250) — compile-verified
//
#include <hip/hip_runtime.h>

typedef __attribute__((ext_vector_type(2))) float v2f;
typedef __attribute__((ext_vector_type(8))) float v8f;

#define N_ 50000
#define E_ 800000
#define G_ 64
#define D_ 128
#define BN_EPS 1e-5f

// ---------------- degree / dinv ----------------
__global__ void deg_init_kernel(float* deg) {
    int i = blockIdx.x * blockDim.x + threadIdx.x;
    if (i < N_) deg[i] = 1.0f;  // self-loop
}

__global__ void deg_acc_kernel(const int* __restrict__ ei, float* deg) {
    int e = blockIdx.x * blockDim.x + threadIdx.x;
    if (e < E_) atomicAdd(&deg[ei[E_ + e]], 1.0f);  // dst row of edge_index
}

__global__ void rsqrt_inplace_kernel(float* deg) {
    int i = blockIdx.x * blockDim.x + threadIdx.x;
    if (i < N_) deg[i] = rsqrtf(deg[i]);
}

// ---------------- WMMA f32 GEMM: H = X @ W  (N x 128 @ 128 x 128) ----------------
// One wave -> one 16x16 output tile. Two accumulator chains for ILP.
__global__ __launch_bounds__(256) void gemm_kernel(const float* __restrict__ X,
                                                   const float* __restrict__ W,
                                                   float* __restrict__ H) {
    const int lane = threadIdx.x & 31;
    const int wave = threadIdx.x >> 5;       // 8 waves -> 8 column tiles (128 cols)
    const int row0 = blockIdx.x * 16;
    const int col0 = wave * 16;
    const int half = lane >> 4;              // 0: K pair {k,k+1}, 1: {k+2,k+3}
    const int l15  = lane & 15;

    const float* xrow = X + (size_t)(row0 + l15) * D_;

    v8f acc0 = {};
    v8f acc1 = {};

#pragma unroll
    for (int k0 = 0; k0 < D_; k0 += 8) {
        // chain 0: K block [k0, k0+4)
        {
            const int ka = k0 + 2 * half;
            v2f a = *(const v2f*)(xrow + ka);                 // A 16x4 fragment
            v2f b;
            b.x = W[(size_t)ka * D_ + col0 + l15];            // B 4x16 fragment
            b.y = W[(size_t)(ka + 1) * D_ + col0 + l15];
            acc0 = __builtin_amdgcn_wmma_f32_16x16x4_f32(
                false, a, false, b, (short)0, acc0, false, false);
        }
        // chain 1: K block [k0+4, k0+8)
        {
            const int kb = k0 + 4 + 2 * half;
            v2f a = *(const v2f*)(xrow + kb);
            v2f b;
            b.x = W[(size_t)kb * D_ + col0 + l15];
            b.y = W[(size_t)(kb + 1) * D_ + col0 + l15];
            acc1 = __builtin_amdgcn_wmma_f32_16x16x4_f32(
                false, a, false, b, (short)0, acc1, false, false);
        }
    }

    v8f acc = acc0 + acc1;

    // C/D layout: VGPR j -> row (j + 8*half), col = l15
#pragma unroll
    for (int j = 0; j < 8; ++j) {
        int row = row0 + j + 8 * half;
        H[(size_t)row * D_ + col0 + l15] = acc[j];
    }
}

// ---------------- agg init: self-loop contribution + bias ----------------
__global__ void node_init_kernel(const float* __restrict__ h,
                                 const float* __restrict__ dinv,
                                 const float* __restrict__ bias,
                                 float* __restrict__ agg) {
    int idx = blockIdx.x * blockDim.x + threadIdx.x;  // N_*D_ threads exactly
    int row = idx >> 7;
    int col = idx & 127;
    float dv = dinv[row];
    agg[idx] = h[idx] * (dv * dv) + bias[col];
}

// ---------------- edge scatter: one wave per edge, float4 per lane ----------------
__global__ __launch_bounds__(256) void edge_agg_kernel(const int* __restrict__ ei,
                                                       const float* __restrict__ dinv,
                                                       const float* __restrict__ h,
                                                       float* __restrict__ agg) {
    int wid  = (blockIdx.x * blockDim.x + threadIdx.x) >> 5;  // edge id
    int lane = threadIdx.x & 31;
    if (wid >= E_) return;
    int s = ei[wid];
    int d = ei[E_ + wid];
    float coef = dinv[s] * dinv[d];
    const float4 v = *(const float4*)(h + (size_t)s * D_ + lane * 4);
    float* dst = agg + (size_t)d * D_ + lane * 4;
    atomicAdd(dst + 0, v.x * coef);
    atomicAdd(dst + 1, v.y * coef);
    atomicAdd(dst + 2, v.z * coef);
    atomicAdd(dst + 3, v.w * coef);
}

// ---------------- zero BN accumulators ----------------
__global__ void zero_sums_kernel(float* gsum, float* gsq) {
    int t = threadIdx.x;
    if (t < 128) gsum[t] = 0.0f;
    else         gsq[t - 128] = 0.0f;
}

// ---------------- in-place ReLU + BN statistics ----------------
__global__ void relu_stats_kernel(float* __restrict__ agg,
                                  float* __restrict__ gsum,
                                  float* __restrict__ gsq) {
    int t   = threadIdx.x;
    int col = t & 127;
    int rg  = t >> 7;          // 0 or 1
    int row0 = blockIdx.x * 64;
    float s = 0.0f, s2 = 0.0f;
    for (int r = rg; r < 64; r += 2) {
        int row = row0 + r;
        if (row >= N_) break;
        int idx = row * D_ + col;
        float v = agg[idx];
        v = fmaxf(v, 0.0f);
        agg[idx] = v;
        s  += v;
        s2 += v * v;
    }
    __shared__ float ls[256];
    __shared__ float lq[256];
    ls[t] = s; lq[t] = s2;
    __syncthreads();
    if (rg == 0) {
        atomicAdd(&gsum[col], ls[t] + ls[t + 128]);
        atomicAdd(&gsq[col],  lq[t] + lq[t + 128]);
    }
}

// ---------------- BN apply ----------------
__global__ void bn_apply_kernel(const float* __restrict__ r,
                                const float* __restrict__ gsum,
                                const float* __restrict__ gsq,
                                const float* __restrict__ gamma,
                                const float* __restrict__ beta,
                                float* __restrict__ xout) {
    int idx = blockIdx.x * blockDim.x + threadIdx.x;  // N_*D_ threads exactly
    int col = idx & 127;
    const float inv_n = 1.0f / (float)N_;
    float mu  = gsum[col] * inv_n;
    float var = gsq[col] * inv_n - mu * mu;
    xout[idx] = (r[idx] - mu) * rsqrtf(var + BN_EPS) * gamma[col] + beta[col];
}

// ---------------- output zero + pooled segment-sum ----------------
__global__ void zero_out_kernel(float* out) {
    int i = blockIdx.x * blockDim.x + threadIdx.x;
    if (i < G_ * D_) out[i] = 0.0f;
}

__global__ __launch_bounds__(256) void pool_kernel(const float* __restrict__ x,
                                                   const int* __restrict__ batch,
                                                   float* __restrict__ out) {
    __shared__ float acc[G_ * D_];  // 32 KB
    int t = threadIdx.x;
    for (int i = t; i < G_ * D_; i += 256) acc[i] = 0.0f;
    __syncthreads();
    int col = t & 127;
    int rg  = t >> 7;
    for (int row = blockIdx.x * 2 + rg; row < N_; row += gridDim.x * 2) {
        int g = batch[row];
        atomicAdd(&acc[g * D_ + col], x[(size_t)row * D_ + col]);
    }
    __syncthreads();
    for (int i = t; i < G_ * D_; i += 256) atomicAdd(&out[i], acc[i]);
}

// ---------------- host launcher ----------------
extern "C" void kernel_launch(void* const* d_in, const int* in_sizes, int n_in,
                              void* d_out, int out_size, void* d_ws, size_t ws_size,
                              hipStream_t stream) {
    const float* x0    = (const float*)d_in[0];
    const int*   ei    = (const int*)d_in[1];
    const int*   batch = (const int*)d_in[2];
    const float* Ws[4]  = {(const float*)d_in[3],  (const float*)d_in[7],
                           (const float*)d_in[11], (const float*)d_in[15]};
    const float* bs[4]  = {(const float*)d_in[4],  (const float*)d_in[8],
                           (const float*)d_in[12], (const float*)d_in[16]};
    const float* gms[4] = {(const float*)d_in[5],  (const float*)d_in[9],
                           (const float*)d_in[13], (const float*)d_in[17]};
    const float* bes[4] = {(const float*)d_in[6],  (const float*)d_in[10],
                           (const float*)d_in[14], (const float*)d_in[18]};

    char* ws = (char*)d_ws;
    const size_t MAT = (size_t)N_ * D_ * sizeof(float);  // 25,600,000 B
    float* h    = (float*)(ws);
    float* agg  = (float*)(ws + MAT);
    float* xb   = (float*)(ws + 2 * MAT);
    float* dinv = (float*)(ws + 3 * MAT);                 // N_ floats
    float* gsum = (float*)(ws + 3 * MAT + 204800);        // 128 floats
    float* gsq  = (float*)(ws + 3 * MAT + 205312);        // 128 floats

    // symmetric-normalization coefficients (recomputed every call; deterministic)
    deg_init_kernel<<<(N_ + 255) / 256, 256, 0, stream>>>(dinv);
    deg_acc_kernel<<<(E_ + 255) / 256, 256, 0, stream>>>(ei, dinv);
    rsqrt_inplace_kernel<<<(N_ + 255) / 256, 256, 0, stream>>>(dinv);

    const float* xin = x0;
    for (int l = 0; l < 4; ++l) {
        zero_sums_kernel<<<1, 256, 0, stream>>>(gsum, gsq);
        gemm_kernel<<<N_ / 16, 256, 0, stream>>>(xin, Ws[l], h);
        node_init_kernel<<<(N_ * D_) / 256, 256, 0, stream>>>(h, dinv, bs[l], agg);
        edge_agg_kernel<<<E_ / 8, 256, 0, stream>>>(ei, dinv, h, agg);
        relu_stats_kernel<<<(N_ + 63) / 64, 256, 0, stream>>>(agg, gsum, gsq);
        bn_apply_kernel<<<(N_ * D_) / 256, 256, 0, stream>>>(agg, gsum, gsq,
                                                             gms[l], bes[l], xb);
        xin = xb;
    }

    zero_out_kernel<<<(G_ * D_ + 255) / 256, 256, 0, stream>>>((float*)d_out);
    pool_kernel<<<128, 256, 0, stream>>>(xb, batch, (float*)d_out);
}